// LiSPNetx22_64991445123206
// MI455X (gfx1250) — compile-verified
//
#include <hip/hip_runtime.h>

#define C   256
#define H   64
#define W   64
#define HW  4096
#define KP  16      // padded EM-basis count
#define KV  8       // actual EM bases
#define NB  4       // batch
#define TILES 32    // K1 position tiles per batch (128 positions each)

typedef __attribute__((ext_vector_type(2))) float v2f;
typedef __attribute__((ext_vector_type(8))) float v8f;

// D = A(16x4 f32) * B(4x16 f32) + C  -> v_wmma_f32_16x16x4_f32
__device__ __forceinline__ v8f wmma4(v2f a, v2f b, v8f c) {
  return __builtin_amdgcn_wmma_f32_16x16x4_f32(false, a, false, b, (short)0, c,
                                               false, false);
}

// ---- broadcast mu0 (1,C,8) -> mu (4,C,16) with zero pad -----------------
__global__ void k_mu_init(const float* __restrict__ mu0, float* __restrict__ mu) {
  int i = blockIdx.x * 256 + threadIdx.x;          // NB*C*KP = 16384
  if (i >= NB * C * KP) return;
  int n = i & 15;
  int c = (i >> 4) & 255;
  mu[i] = (n < KV) ? mu0[c * KV + n] : 0.f;
}

// ---- one EM stage: z = softmax(x^T mu); partial colsum & mu numerator ----
// grid: (32 tiles, 4 batch); block 256 = 8 waves, each wave owns 16 positions
__global__ __launch_bounds__(256)
void k_em_stage(const float* __restrict__ x,      // [4][256][4096]
                const float* __restrict__ mu,     // [4][256][16]
                float* __restrict__ npart,        // [32][4][256*16]
                float* __restrict__ cspart,       // [32][4][16]
                float* __restrict__ zb,           // [4][4096][16] (this scale)
                int size, int pad)
{
  __shared__ float zw[128 * KP];   // W-weighted z for the block's 128 positions
  __shared__ float csw[8 * KV];    // per-wave colsum partials

  const int tid  = threadIdx.x;
  const int lane = tid & 31;
  const int wid  = tid >> 5;
  const int tile = blockIdx.x;
  const int bb   = blockIdx.y;
  const int pos0 = tile * 128 + wid * 16;
  const int ln   = lane & 15;
  const int lh   = lane >> 4;

  const float* xb  = x  + bb * (C * HW);
  const float* mub = mu + bb * (C * KP);

  // GEMM1: S(16 pos x 16 k) = x^T * mu, inner dim = 256 channels
  v8f s = {};
  for (int k0 = 0; k0 < C; k0 += 4) {
    int cA = k0 + lh * 2;
    v2f a, b;
    a.x = xb[(cA + 0) * HW + pos0 + ln];
    a.y = xb[(cA + 1) * HW + pos0 + ln];
    b.x = mub[(cA + 0) * KP + ln];
    b.y = mub[(cA + 1) * KP + ln];
    s = wmma4(a, b, s);
  }

  // softmax over k (lane dim, valid k = lanes 0..7 of each half)
  float csl = 0.f;
  for (int r = 0; r < 8; ++r) {
    float v = s[r];
    float m = v;
    m = fmaxf(m, __shfl_xor(m, 1, 32));
    m = fmaxf(m, __shfl_xor(m, 2, 32));
    m = fmaxf(m, __shfl_xor(m, 4, 32));
    float e = __expf(v - m);
    float sum = e;
    sum += __shfl_xor(sum, 1, 32);
    sum += __shfl_xor(sum, 2, 32);
    sum += __shfl_xor(sum, 4, 32);
    float z = e / sum;

    int mrow = r + lh * 8;                 // position offset in tile
    int pos  = pos0 + mrow;
    int y = pos >> 6, xc = pos & 63;
    int wy = min(size - 1, y + pad)  - max(0, y + pad  - (H - 1)) + 1;
    int wx = min(size - 1, xc + pad) - max(0, xc + pad - (W - 1)) + 1;
    float wgt = (float)(wy * wx);          // unfold/fold multiplicity
    float zz  = (ln < KV) ? z : 0.f;
    zw[(wid * 16 + mrow) * KP + ln]  = zz * wgt;
    zb[(bb * HW + pos) * KP + ln]    = zz;       // raw z for reconstruction
    csl += zz * wgt;
  }
  csl += __shfl_xor(csl, 16, 32);
  if (lane < KV) csw[wid * KV + lane] = csl;
  __syncthreads();

  // GEMM2: npart[c,k] = sum_p x[c,p] * zw[p,k] over the block's 128 positions
  float* np = npart + (tile * NB + bb) * (C * KP);
  for (int half = 0; half < 2; ++half) {
    int c0 = (wid + half * 8) * 16;        // each wave owns 2 disjoint c-tiles
    v8f d = {};
    for (int ks = 0; ks < 32; ++ks) {
      int pk = ks * 4 + lh * 2;
      v2f a, b;
      a.x = xb[(c0 + ln) * HW + tile * 128 + pk + 0];
      a.y = xb[(c0 + ln) * HW + tile * 128 + pk + 1];
      b.x = zw[(pk + 0) * KP + ln];
      b.y = zw[(pk + 1) * KP + ln];
      d = wmma4(a, b, d);
    }
    for (int r = 0; r < 8; ++r)
      np[(c0 + r + lh * 8) * KP + ln] = d[r];
  }

  if (tid < KV) {
    float t = 0.f;
    for (int w8 = 0; w8 < 8; ++w8) t += csw[w8 * KV + tid];
    cspart[(tile * NB + bb) * KP + tid] = t;
  }
}

// ---- reduce partials, divide by (1e-6+colsum), l2-normalize mu over c ----
__global__ __launch_bounds__(256)
void k_mu_update(const float* __restrict__ npart,
                 const float* __restrict__ cspart,
                 float* __restrict__ mu,
                 float oob8)                       // (#OOB rows)/8 per scale
{
  __shared__ float cs[KV];
  __shared__ float val[C * KV];
  __shared__ float nrm[KV];
  int bb = blockIdx.x;
  int c  = threadIdx.x;

  if (c < KV) {
    float t = oob8;
    for (int tile = 0; tile < TILES; ++tile)
      t += cspart[(tile * NB + bb) * KP + c];
    cs[c] = 1e-6f + t;
  }
  __syncthreads();

  for (int n = 0; n < KV; ++n) {
    float a = 0.f;
    for (int tile = 0; tile < TILES; ++tile)
      a += npart[(tile * NB + bb) * (C * KP) + c * KP + n];
    val[c * KV + n] = a / cs[n];
  }
  __syncthreads();

  if (c < KV) {
    float ss = 0.f;
    for (int i = 0; i < C; ++i) { float v = val[i * KV + c]; ss += v * v; }
    nrm[c] = 1e-6f + sqrtf(ss);
  }
  __syncthreads();

  float* mub = mu + bb * C * KP + c * KP;
  for (int n = 0; n < KP; ++n)
    mub[n] = (n < KV) ? (val[c * KV + n] / nrm[n]) : 0.f;
}

// ---- G_s = convcat_slice^T * mu_s : (256 x 16), folds 1x1 conv into k-space
__global__ __launch_bounds__(256)
void k_gmat(const float* __restrict__ cw,   // [256][768]
            const float* __restrict__ mu,   // [4][256][16]
            float* __restrict__ g,          // [4][256][16] (this scale)
            int coff)
{
  int lane = threadIdx.x & 31, wid = threadIdx.x >> 5;
  int o0 = (blockIdx.x * 8 + wid) * 16;
  int bb = blockIdx.y;
  int ln = lane & 15, lh = lane >> 4;
  const float* mub = mu + bb * C * KP;
  v8f d = {};
  for (int k0 = 0; k0 < C; k0 += 4) {
    int i = k0 + lh * 2;
    v2f a, b;
    a.x = cw[(o0 + ln) * (3 * C) + coff + i + 0];
    a.y = cw[(o0 + ln) * (3 * C) + coff + i + 1];
    b.x = mub[(i + 0) * KP + ln];
    b.y = mub[(i + 1) * KP + ln];
    d = wmma4(a, b, d);
  }
  float* gb = g + bb * C * KP;
  for (int r = 0; r < 8; ++r)
    gb[(o0 + r + lh * 8) * KP + ln] = d[r];
}

// ---- out = relu(w_scale * (sum_s W_s(p) * (G_s Z_s^T) + bias) + x) -------
__global__ __launch_bounds__(256)
void k_output(const float* __restrict__ x,
              const float* __restrict__ g,    // [3][4][256][16]
              const float* __restrict__ zb,   // [3][4][4096][16]
              const float* __restrict__ cb,   // [256]
              const float* __restrict__ wsc,  // scalar
              float* __restrict__ out)
{
  int lane = threadIdx.x & 31, wid = threadIdx.x >> 5;
  int ln = lane & 15, lh = lane >> 4;
  int p0 = blockIdx.x * 16;
  int o0 = (blockIdx.y * 8 + wid) * 16;
  int bb = blockIdx.z;

  int pos = p0 + ln;
  int y = pos >> 6, xc = pos & 63;

  v8f acc = {};
  for (int si = 0; si < 3; ++si) {
    int size = 2 * si + 1, pad = si;
    const float* gs = g  + (si * NB + bb) * (C * KP);
    const float* zs = zb + (size_t)(si * NB + bb) * (HW * KP);
    v8f t = {};
    for (int ks = 0; ks < 4; ++ks) {
      int kk = ks * 4 + lh * 2;
      v2f a, b;
      a.x = gs[(o0 + ln) * KP + kk + 0];
      a.y = gs[(o0 + ln) * KP + kk + 1];
      b.x = zs[pos * KP + kk + 0];
      b.y = zs[pos * KP + kk + 1];
      t = wmma4(a, b, t);
    }
    int wy = min(size - 1, y + pad)  - max(0, y + pad  - (H - 1)) + 1;
    int wx = min(size - 1, xc + pad) - max(0, xc + pad - (W - 1)) + 1;
    float wgt = (float)(wy * wx);
    acc += t * wgt;
  }

  float ws = wsc[0];
  const float* xb = x   + (size_t)bb * C * HW;
  float*       ob = out + (size_t)bb * C * HW;
  for (int r = 0; r < 8; ++r) {
    int o = o0 + r + lh * 8;
    float v = ws * (acc[r] + cb[o]) + xb[o * HW + pos];
    ob[o * HW + pos] = fmaxf(v, 0.f);
  }
}

// ---- emit final mu (4,256,8) into the output tail ------------------------
__global__ void k_mu_out(const float* __restrict__ mu, float* __restrict__ om) {
  int i = blockIdx.x * 256 + threadIdx.x;   // NB*C*KV = 8192
  if (i >= NB * C * KV) return;
  om[i] = mu[(i >> 3) * KP + (i & 7)];
}

extern "C" void kernel_launch(void* const* d_in, const int* in_sizes, int n_in,
                              void* d_out, int out_size, void* d_ws, size_t ws_size,
                              hipStream_t stream) {
  const float* x   = (const float*)d_in[0];
  const float* mu0 = (const float*)d_in[1];
  const float* wsc = (const float*)d_in[2];
  const float* cw  = (const float*)d_in[3];
  const float* cb  = (const float*)d_in[4];
  float* out = (float*)d_out;

  float* ws     = (float*)d_ws;
  float* mu     = ws;                               // 4*256*16      = 16384
  float* npart  = mu     + NB * C * KP;             // 32*4*256*16   = 524288
  float* cspart = npart  + TILES * NB * C * KP;     // 32*4*16       = 2048
  float* g      = cspart + TILES * NB * KP;         // 3*4*256*16    = 49152
  float* zb     = g      + 3 * NB * C * KP;         // 3*4*4096*16   = 786432

  k_mu_init<<<dim3(64), dim3(256), 0, stream>>>(mu0, mu);

  // #rows with all-zero unfold columns, divided by 8 (uniform softmax mass)
  const float oob8[3] = {0.f, 764.f / 8.f, 3804.f / 8.f};
  for (int si = 0; si < 3; ++si) {
    int size = 2 * si + 1, pad = si;
    float* zbs = zb + (size_t)si * NB * HW * KP;
    for (int st = 0; st < 7; ++st) {
      k_em_stage<<<dim3(TILES, NB), dim3(256), 0, stream>>>(
          x, mu, npart, cspart, zbs, size, pad);
      k_mu_update<<<dim3(NB), dim3(256), 0, stream>>>(
          npart, cspart, mu, oob8[si]);
    }
    k_gmat<<<dim3(2, NB), dim3(256), 0, stream>>>(
        cw, mu, g + (size_t)si * NB * C * KP, si * C);
  }

  k_output<<<dim3(HW / 16, 2, NB), dim3(256), 0, stream>>>(
      x, g, zb, cb, wsc, out);
  k_mu_out<<<dim3(32), dim3(256), 0, stream>>>(
      mu, out + (size_t)NB * C * HW);
}